// GraphSageNet1_83073257439659
// MI455X (gfx1250) — compile-verified
//
#include <hip/hip_runtime.h>
#include <hip/hip_bf16.h>

typedef float v2f __attribute__((ext_vector_type(2)));
typedef float v8f __attribute__((ext_vector_type(8)));

// ---------------------------------------------------------------------------
// WMMA f32 GEMM:  OUT[M,Ncols] = concatK(A1[M,K1], A2[M,K-K1]*rdeg) @ W[K,Ncols] + bias
//
// Block = 128 threads = 4 waves. blockIdx.y = column tile (shared by all 4
// waves -> one LDS-staged W panel per block), blockIdx.x*4+wid = row tile.
// Each wave computes a 16x16 tile via V_WMMA_F32_16X16X4_F32, K stepped by 4.
// A frag = one global_load_b64 (row-major adjacent), B frag = one ds_load_b64
// from the transposed LDS panel. K/K1 are compile-time -> no divergent concat
// branch, constant strides. M must be a multiple of 16 (N=100000 = 6250*16).
// ---------------------------------------------------------------------------
template <int K, int K1, bool RELU>
__global__ __launch_bounds__(128)
void k_gemm_wmma(const float* __restrict__ A1, const float* __restrict__ A2,
                 const float* __restrict__ rdeg,
                 const float* __restrict__ W, const float* __restrict__ bias,
                 float* __restrict__ OUT, int M, int Ncols)
{
    constexpr int K2 = K - K1;
    constexpr int Kp = K + 2;                 // pad to spread LDS banks
    __shared__ float smemB[16 * Kp];

    const int col0 = blockIdx.y << 4;

    // Cooperative load of W[:, col0:col0+16] into LDS, transposed: smemB[c][kk].
    // Consecutive threads read consecutive columns -> coalesced 64B segments.
    for (int idx = threadIdx.x; idx < 16 * K; idx += 128) {
        const int c  = idx & 15;
        const int kk = idx >> 4;
        const int col = col0 + c;
        smemB[c * Kp + kk] = (col < Ncols) ? W[(size_t)kk * Ncols + col] : 0.f;
    }
    __syncthreads();

    const int wid  = threadIdx.x >> 5;
    const int lane = threadIdx.x & 31;
    const int tm   = blockIdx.x * 4 + wid;
    if (tm >= (M >> 4)) return;               // wave-uniform exit (EXEC stays full)

    const int lr   = lane & 15;               // A row / B,D col within tile
    const int kg   = (lane >> 4) << 1;        // K sub-offset {0,2}
    const int arow = (tm << 4) + lr;
    const int col  = col0 + lr;

    const float* __restrict__ a1p = A1 + (size_t)arow * K1 + kg;
    const v2f*   __restrict__ bp  = (const v2f*)(smemB + lr * Kp + kg);

    v8f acc = {0.f, 0.f, 0.f, 0.f, 0.f, 0.f, 0.f, 0.f};

    for (int k = 0; k < K1; k += 4) {
        v2f a = *(const v2f*)(a1p + k);       // global_load_b64
        v2f b = bp[k >> 1];                   // ds_load_b64
        acc = __builtin_amdgcn_wmma_f32_16x16x4_f32(
                  false, a, false, b, (short)0, acc, false, false);
    }
    if constexpr (K2 > 0) {
        const float r = rdeg[arow];           // fold mean-aggregation scale into A2
        const float* __restrict__ a2p = A2 + (size_t)arow * K2 + kg;
        for (int k = 0; k < K2; k += 4) {
            v2f araw = *(const v2f*)(a2p + k);
            v2f a; a.x = araw.x * r; a.y = araw.y * r;
            v2f b = bp[(K1 + k) >> 1];
            acc = __builtin_amdgcn_wmma_f32_16x16x4_f32(
                      false, a, false, b, (short)0, acc, false, false);
        }
    }

    if (col < Ncols) {
        const float bs = bias[col];
        // D layout: VGPR r -> row = 16*tm + r + 8*(lane/16)
        float* op = OUT + (size_t)((tm << 4) + ((lane >> 4) << 3)) * Ncols + col;
        #pragma unroll
        for (int r8 = 0; r8 < 8; ++r8) {
            float v = acc[r8] + bs;
            if constexpr (RELU) v = fmaxf(v, 0.f);
            op[(size_t)r8 * Ncols] = v;
        }
    }
}

// -------------------- graph / elementwise kernels (H = 108 = 27 float4) ----

#define HDIM 108
#define H4   27

__global__ void k_degree(const int* __restrict__ dst, float* __restrict__ deg, int E) {
    int e = blockIdx.x * blockDim.x + threadIdx.x;
    if (e < E) atomicAdd(&deg[dst[e]], 1.0f);
}

__global__ void k_recip_clamp(float* __restrict__ deg, int N) {
    int n = blockIdx.x * blockDim.x + threadIdx.x;
    if (n < N) deg[n] = 1.0f / fmaxf(deg[n], 1.0f);
}

// wave per edge: c[dst] += z[src]; lane j handles float4 j (27 active lanes)
__global__ void k_scatter(const float* __restrict__ z,
                          const int* __restrict__ src, const int* __restrict__ dst,
                          float* __restrict__ c, int E)
{
    int t = blockIdx.x * blockDim.x + threadIdx.x;
    int e = t >> 5;
    int lane = t & 31;
    if (e >= E || lane >= H4) return;
    const float4 v = ((const float4*)(z + (size_t)src[e] * HDIM))[lane];
    float* cr = c + (size_t)dst[e] * HDIM + lane * 4;
    atomicAdd(cr + 0, v.x);
    atomicAdd(cr + 1, v.y);
    atomicAdd(cr + 2, v.z);
    atomicAdd(cr + 3, v.w);
}

__device__ __forceinline__ float wave_sum(float v) {
    #pragma unroll
    for (int off = 16; off > 0; off >>= 1)
        v += __shfl_xor(v, off, 32);
    return v;
}

// wave per node: h += relu(b / max(||b||, 1e-12))
__global__ void k_norm_residual(const float* __restrict__ b, float* __restrict__ h, int N)
{
    int t = blockIdx.x * blockDim.x + threadIdx.x;
    int n = t >> 5;
    int lane = t & 31;
    if (n >= N) return;
    const float4* br = (const float4*)(b + (size_t)n * HDIM);
    float4*       hr = (float4*)(h + (size_t)n * HDIM);
    float4 v = {0.f, 0.f, 0.f, 0.f};
    float ss = 0.f;
    if (lane < H4) {
        v = br[lane];
        ss = v.x * v.x + v.y * v.y + v.z * v.z + v.w * v.w;
    }
    ss = wave_sum(ss);
    const float inv = 1.0f / fmaxf(sqrtf(ss), 1e-12f);
    if (lane < H4) {
        float4 hv = hr[lane];
        hv.x += fmaxf(v.x * inv, 0.f);
        hv.y += fmaxf(v.y * inv, 0.f);
        hv.z += fmaxf(v.z * inv, 0.f);
        hv.w += fmaxf(v.w * inv, 0.f);
        hr[lane] = hv;
    }
}

__global__ void k_count_nodes(const int* __restrict__ gids, float* __restrict__ cnt, int N) {
    int n = blockIdx.x * blockDim.x + threadIdx.x;
    if (n < N) atomicAdd(&cnt[gids[n]], 1.0f);
}

// wave per node: out[gid] += h[n]
__global__ void k_readout_acc(const float* __restrict__ h, const int* __restrict__ gids,
                              float* __restrict__ out, int N)
{
    int t = blockIdx.x * blockDim.x + threadIdx.x;
    int n = t >> 5;
    int lane = t & 31;
    if (n >= N || lane >= H4) return;
    const float4 v = ((const float4*)(h + (size_t)n * HDIM))[lane];
    float* og = out + (size_t)gids[n] * HDIM + lane * 4;
    atomicAdd(og + 0, v.x);
    atomicAdd(og + 1, v.y);
    atomicAdd(og + 2, v.z);
    atomicAdd(og + 3, v.w);
}

__global__ void k_readout_div(float* __restrict__ out, const float* __restrict__ cnt,
                              int G)
{
    int i = blockIdx.x * blockDim.x + threadIdx.x;
    if (i < G * HDIM) out[i] /= fmaxf(cnt[i / HDIM], 1.0f);
}

// ---------------------------------------------------------------------------

static inline int cdiv(long long a, long long b) { return (int)((a + b - 1) / b); }

extern "C" void kernel_launch(void* const* d_in, const int* in_sizes, int n_in,
                              void* d_out, int out_size, void* d_ws, size_t ws_size,
                              hipStream_t stream)
{
    const float* nodes_feat = (const float*)d_in[0];
    const float* W_emb      = (const float*)d_in[4];
    const float* b_emb      = (const float*)d_in[5];
    const float* pool_W     = (const float*)d_in[6];
    const float* pool_b     = (const float*)d_in[7];
    const float* app_W      = (const float*)d_in[8];
    const float* app_b      = (const float*)d_in[9];
    const int*   src        = (const int*)d_in[10];
    const int*   dst        = (const int*)d_in[11];
    const int*   gids       = (const int*)d_in[12];

    const int N  = in_sizes[2];              // nodes_num_norm_sqrt: (N,) -> 100000
    const int E  = in_sizes[10];             // 1.6M
    const int H  = HDIM;                     // 108 (in_sizes[5])
    const int G  = out_size / H;             // 128
    const int NL = in_sizes[7] / H;          // 2 layers

    // workspace carve-out: 3*N*H + N + G floats (~130 MB)
    float* h    = (float*)d_ws;
    float* z    = h + (size_t)N * H;         // reused for app-GEMM output b
    float* c    = z + (size_t)N * H;
    float* rdeg = c + (size_t)N * H;
    float* cnt  = rdeg + N;
    float* out  = (float*)d_out;

    const int TPB = 256;
    const int mtiles  = N / 16;
    const int ntilesN = (H + 15) / 16;       // 7
    const dim3 gemmGrid(cdiv(mtiles, 4), ntilesN);

    // init
    hipMemsetAsync(rdeg, 0, (size_t)N * sizeof(float), stream);
    hipMemsetAsync(cnt,  0, (size_t)G * sizeof(float), stream);
    hipMemsetAsync(out,  0, (size_t)G * H * sizeof(float), stream);

    // in-degree -> 1/max(deg,1)
    k_degree<<<cdiv(E, TPB), TPB, 0, stream>>>(dst, rdeg, E);
    k_recip_clamp<<<cdiv(N, TPB), TPB, 0, stream>>>(rdeg, N);

    // h = nodes_feat @ W_emb + b_emb           (K = IN = 64)
    k_gemm_wmma<64, 64, false><<<gemmGrid, 128, 0, stream>>>(
        nodes_feat, nullptr, nullptr, W_emb, b_emb, h, N, H);

    for (int l = 0; l < NL; ++l) {
        const float* pW = pool_W + (size_t)l * H * H;
        const float* pb = pool_b + (size_t)l * H;
        const float* aW = app_W  + (size_t)l * 2 * H * H;
        const float* ab = app_b  + (size_t)l * H;

        // z = relu(h @ pool_W + pool_b)        (K = 108)
        k_gemm_wmma<108, 108, true><<<gemmGrid, 128, 0, stream>>>(
            h, nullptr, nullptr, pW, pb, z, N, H);

        // c = segment_sum(z[src] -> dst)
        hipMemsetAsync(c, 0, (size_t)N * H * sizeof(float), stream);
        k_scatter<<<cdiv((long long)E * 32, TPB), TPB, 0, stream>>>(z, src, dst, c, E);

        // b = concat(h, c/deg) @ app_W + app_b (K = 216, split at 108; deg scale fused)
        k_gemm_wmma<216, 108, false><<<gemmGrid, 128, 0, stream>>>(
            h, c, rdeg, aW, ab, z, N, H);

        // h = h + relu(b / max(||b||, 1e-12))
        k_norm_residual<<<cdiv((long long)N * 32, TPB), TPB, 0, stream>>>(z, h, N);
    }

    // per-graph mean readout
    k_count_nodes<<<cdiv(N, TPB), TPB, 0, stream>>>(gids, cnt, N);
    k_readout_acc<<<cdiv((long long)N * 32, TPB), TPB, 0, stream>>>(h, gids, out, N);
    k_readout_div<<<cdiv((long long)G * H, TPB), TPB, 0, stream>>>(out, cnt, G);
}